// QuantizeEMAChannelWise_39041252720884
// MI455X (gfx1250) — compile-verified
//
#include <hip/hip_runtime.h>
#include <stdint.h>

typedef __attribute__((ext_vector_type(2))) float v2f;
typedef __attribute__((ext_vector_type(4))) float v4f;
typedef __attribute__((ext_vector_type(8))) float v8f;

#define NROWS   32768        // B*C = 128*256
#define HW      256          // H*W
#define NCODE   1024         // codewords
#define KSTEPS  64           // HW / 4  (K per wmma = 4)
#define NTILES  64           // NCODE / 16
#define ROWS_PER_BLOCK 128   // 8 waves * 16 rows
#define BPAD    260          // padded LDS row stride (floats): rows 4 banks apart
#define GAMMA   0.99f

// d_ws layout (floats): [0,1024) cnorm | [1024,2048) counts | [2048, 2048+262144) sums

__global__ void vq_prep(const float* __restrict__ cb, float* __restrict__ ws) {
    int k = blockIdx.x * blockDim.x + threadIdx.x;   // 0..1023
    if (k >= NCODE) return;
    const float* c = cb + k * HW;
    float* sums = ws + 2048;
    float s = 0.f;
    v4f z = {0.f, 0.f, 0.f, 0.f};
    for (int j = 0; j < HW; j += 4) {
        v4f v = *(const v4f*)(c + j);
        s += v.x * v.x + v.y * v.y + v.z * v.z + v.w * v.w;
        *(v4f*)(sums + k * HW + j) = z;
    }
    ws[k] = s;            // cnorm
    ws[1024 + k] = 0.f;   // counts
}

// Issue this thread's 4 x 16B async global->LDS copies for one 16x256 B tile.
__device__ __forceinline__ void stage_tile_async(const float* __restrict__ cb,
                                                 int n0, float* dstbase, int tid) {
#pragma unroll
    for (int j = 0; j < 4; ++j) {
        int u  = tid + j * 256;             // float4 unit 0..1023
        int r  = u >> 6;                    // codeword within tile
        int c4 = (u & 63) << 2;             // float offset in row
        unsigned           lds_addr = (unsigned)(uintptr_t)(dstbase + r * BPAD + c4);
        unsigned long long gaddr    = (unsigned long long)(uintptr_t)(cb + (n0 + r) * HW + c4);
        asm volatile("global_load_async_to_lds_b128 %0, %1, off"
                     :: "v"(lds_addr), "v"(gaddr) : "memory");
    }
}

__global__ __launch_bounds__(256) void vq_argmin_wmma(
        const float* __restrict__ x, const float* __restrict__ cb,
        float* __restrict__ ws, float* __restrict__ out) {
    __shared__ float b_lds[2][16 * BPAD];       // double-buffered 16-codeword tiles
    __shared__ int   best_lds[ROWS_PER_BLOCK];

    const int tid  = threadIdx.x;
    const int lane = tid & 31;
    const int wave = tid >> 5;
    const int half = lane >> 4;                 // 0: rows 0-7 / k 0,1 ; 1: rows 8-15 / k 2,3
    const int l15  = lane & 15;

    const float* cnorm  = ws;
    float*       counts = ws + 1024;
    float*       sums   = ws + 2048;

    // ---- A fragments for full K=256 (ISA 16x4 f32 A layout), plus ||x||^2 ----
    const int row  = blockIdx.x * ROWS_PER_BLOCK + wave * 16 + l15;
    const int koff = half * 2;
    const float* xrow = x + row * HW;
    v2f  afrag[KSTEPS];
    float part = 0.f;
#pragma unroll
    for (int kk = 0; kk < KSTEPS; ++kk) {
        v2f a = *(const v2f*)(xrow + 4 * kk + koff);
        afrag[kk] = a;
        part += a.x * a.x + a.y * a.y;
    }
    part += __shfl_xor(part, 16, 32);           // full row norm in lanes r and r+16
    float xn[8];
#pragma unroll
    for (int i = 0; i < 8; ++i)
        xn[i] = __shfl(part, half * 8 + i, 32); // each lane needs rows (half*8 + i)

    float bestv[8];
    int   besti[8];
#pragma unroll
    for (int i = 0; i < 8; ++i) { bestv[i] = 3.4e38f; besti[i] = 0; }

    // ---- prologue: async-stage tile 0 into buffer 0 ----
    stage_tile_async(cb, 0, &b_lds[0][0], tid);

    // ---- N loop: 64 tiles of 16 codewords, double-buffered async staging ----
    for (int nt = 0; nt < NTILES; ++nt) {
        const int cur = nt & 1;
        if (nt + 1 < NTILES) {
            // overlap next tile's global->LDS copy with this tile's WMMAs
            stage_tile_async(cb, (nt + 1) * 16, &b_lds[cur ^ 1][0], tid);
            // async ops complete in order: <=4 outstanding => current buffer done
            asm volatile("s_wait_asynccnt 0x4" ::: "memory");
        } else {
            asm volatile("s_wait_asynccnt 0x0" ::: "memory");
        }
        __syncthreads();                        // all waves' cur-buffer writes visible

        v8f acc = {0.f, 0.f, 0.f, 0.f, 0.f, 0.f, 0.f, 0.f};
        const float* brow = &b_lds[cur][l15 * BPAD + koff];
#pragma unroll
        for (int kc = 0; kc < KSTEPS; kc += 8) {
            v2f bf[8];                          // distinct temps -> clause loads, one wait
#pragma unroll
            for (int j = 0; j < 8; ++j)
                bf[j] = *(const v2f*)(brow + 4 * (kc + j));
#pragma unroll
            for (int j = 0; j < 8; ++j)
                acc = __builtin_amdgcn_wmma_f32_16x16x4_f32(
                          false, afrag[kc + j], false, bf[j], (short)0, acc, false, false);
        }

        // epilogue: d2 = ||x||^2 + ||c||^2 - 2*dot ; running argmin (first-index ties)
        const int   col = nt * 16 + l15;
        const float cn  = cnorm[col];
#pragma unroll
        for (int i = 0; i < 8; ++i) {
            float d2 = xn[i] + cn - 2.f * acc[i];
            if (d2 < bestv[i]) { bestv[i] = d2; besti[i] = col; }
        }
        __syncthreads();                        // all reads of cur done before overwrite
    }

    // ---- argmin butterfly within each 16-lane half ----
#pragma unroll
    for (int off = 1; off <= 8; off <<= 1) {
#pragma unroll
        for (int i = 0; i < 8; ++i) {
            float ov = __shfl_xor(bestv[i], off, 32);
            int   oi = __shfl_xor(besti[i], off, 32);
            if (ov < bestv[i] || (ov == bestv[i] && oi < besti[i])) {
                bestv[i] = ov; besti[i] = oi;
            }
        }
    }
    if (l15 == 0) {
#pragma unroll
        for (int i = 0; i < 8; ++i)
            best_lds[wave * 16 + half * 8 + i] = besti[i];
    }
    __syncthreads();

    // ---- straight-through output + scatter-add (coalesced: thread = column) ----
    const int blkRow0 = blockIdx.x * ROWS_PER_BLOCK;
    for (int r = 0; r < ROWS_PER_BLOCK; ++r) {
        const int idx  = best_lds[r];
        const int grow = blkRow0 + r;
        float xv = x[grow * HW + tid];
        float cv = cb[idx * HW + tid];
        out[grow * HW + tid] = xv + (cv - xv);
        atomicAdd(&sums[idx * HW + tid], xv);
    }
    if (tid < ROWS_PER_BLOCK)
        atomicAdd(&counts[best_lds[tid]], 1.0f);
}

__global__ void vq_ema(const float* __restrict__ Nv, const float* __restrict__ m,
                       const float* __restrict__ ws, float* __restrict__ outN,
                       float* __restrict__ outM, float* __restrict__ outCB) {
    int k = blockIdx.x;       // codeword
    int c = threadIdx.x;      // element
    float cnt  = ws[1024 + k];
    bool  upd  = cnt > 0.f;
    float newN = upd ? (Nv[k] * GAMMA + cnt * (1.f - GAMMA)) : Nv[k];
    if (c == 0) outN[k] = newN;
    float mm = m[k * HW + c];
    float s  = ws[2048 + k * HW + c];
    float nm = upd ? (mm * GAMMA + s * (1.f - GAMMA)) : mm;
    outM[k * HW + c]  = nm;
    outCB[k * HW + c] = nm / newN;
}

extern "C" void kernel_launch(void* const* d_in, const int* in_sizes, int n_in,
                              void* d_out, int out_size, void* d_ws, size_t ws_size,
                              hipStream_t stream) {
    const float* x  = (const float*)d_in[0];   // (128,256,16,16)
    const float* cb = (const float*)d_in[1];   // (1024,1,16,16)
    const float* Nv = (const float*)d_in[2];   // (1024,)
    const float* m  = (const float*)d_in[3];   // (1024,1,16,16)

    float* out   = (float*)d_out;              // (128,256,16,16)
    float* outN  = out  + (size_t)NROWS * HW;  // (1024,)
    float* outM  = outN + NCODE;               // (1024,1,16,16)
    float* outCB = outM + (size_t)NCODE * HW;  // (1024,1,16,16)
    float* ws    = (float*)d_ws;               // cnorm | counts | sums

    vq_prep<<<NCODE / 256, 256, 0, stream>>>(cb, ws);
    vq_argmin_wmma<<<NROWS / ROWS_PER_BLOCK, 256, 0, stream>>>(x, cb, ws, out);
    vq_ema<<<NCODE, 256, 0, stream>>>(Nv, m, ws, outN, outM, outCB);
}